// LMU_general_65592740544900
// MI455X (gfx1250) — compile-verified
//
#include <hip/hip_runtime.h>

typedef __attribute__((ext_vector_type(2))) float v2f;
typedef __attribute__((ext_vector_type(8))) float v8f;

#define NN   49
#define DDm  4
#define INPC 96
#define BSZ  512
#define LEN  1024

#define XSTR 100   // LDS stride for x tile  (100 mod 64 -> conflict-free rows)
#define HSTR 52    // LDS stride for h tiles (52  mod 64 -> conflict-free rows)

// ---- fragment workspace layout (units: float2 == one lane's B-fragment pair) ----
#define OFF_XF0 0        // layer1 x-part: 24 ksteps * 4 ntiles * 32 lanes
#define OFF_XF1 3072     // layers 2-4 x-part: 13*4*32 each
#define OFF_HF0 8064     // Wh per layer: 13*4*32 each
#define OFF_MF0 14720    // Wm per layer: 1*4*32 each
#define OFF_FCF 15232    // fc: 13 ksteps * 6 ntiles * 32 lanes
#define TOT_F2  17728

// ---------------------------------------------------------------------------
// fragment element helpers (B matrix is K x N; W given row-major [N][K])
// ---------------------------------------------------------------------------
__device__ __forceinline__ float wxh_elem(const float* W, const float* e,
                                          int K, int n, int Kdim) {
  if (K >= Kdim || n > 49) return 0.0f;
  if (n < 49) return W[n * Kdim + K];
  return e[K];                       // column 49 carries the ex/eh gate vector
}

__device__ __forceinline__ float fc_elem(const float* fcW, const float* fcb,
                                         int K, int n) {
  if (K < 49)  return fcW[n * 49 + K];
  if (K == 49) return fcb[n];        // bias row (paired with h4 col49 == 1.0)
  return 0.0f;
}

__device__ __forceinline__ void decode(int local, int NT, int& K0, int& n) {
  int lane = local & 31;
  int rem  = local >> 5;
  int nt   = rem % NT;
  int ks   = rem / NT;
  K0 = ks * 4 + ((lane >> 4) << 1);
  n  = nt * 16 + (lane & 15);
}

// ---------------------------------------------------------------------------
// prep: pack all weights into per-lane WMMA B-fragment order in d_ws
// ---------------------------------------------------------------------------
__global__ void lmu_prep(
    const float* ex1, const float* eh1, const float* Wx1, const float* Wh1, const float* Wm1,
    const float* ex2, const float* eh2, const float* Wx2, const float* Wh2, const float* Wm2,
    const float* ex3, const float* eh3, const float* Wx3, const float* Wh3, const float* Wm3,
    const float* ex4, const float* eh4, const float* Wx4, const float* Wh4, const float* Wm4,
    const float* fcW, const float* fcb, float2* __restrict__ ws)
{
  int gid = blockIdx.x * blockDim.x + threadIdx.x;
  if (gid >= TOT_F2) return;
  float a0 = 0.0f, a1 = 0.0f;

  if (gid < OFF_XF1) {                                   // layer-1 x fragments
    int K0, n; decode(gid, 4, K0, n);
    a0 = wxh_elem(Wx1, ex1, K0,     n, 96);
    a1 = wxh_elem(Wx1, ex1, K0 + 1, n, 96);
  } else if (gid < OFF_HF0) {                            // layers 2-4 x fragments
    int li = (gid - OFF_XF1) / 1664, local = (gid - OFF_XF1) % 1664;
    const float* W = (li == 0) ? Wx2 : (li == 1) ? Wx3 : Wx4;
    const float* e = (li == 0) ? ex2 : (li == 1) ? ex3 : ex4;
    int K0, n; decode(local, 4, K0, n);
    a0 = wxh_elem(W, e, K0, n, 49);
    a1 = wxh_elem(W, e, K0 + 1, n, 49);
  } else if (gid < OFF_MF0) {                            // Wh fragments
    int li = (gid - OFF_HF0) / 1664, local = (gid - OFF_HF0) % 1664;
    const float* W = (li == 0) ? Wh1 : (li == 1) ? Wh2 : (li == 2) ? Wh3 : Wh4;
    const float* e = (li == 0) ? eh1 : (li == 1) ? eh2 : (li == 2) ? eh3 : eh4;
    int K0, n; decode(local, 4, K0, n);
    a0 = wxh_elem(W, e, K0, n, 49);
    a1 = wxh_elem(W, e, K0 + 1, n, 49);
  } else if (gid < OFF_FCF) {                            // Wm fragments (K=4)
    int li = (gid - OFF_MF0) / 128, local = (gid - OFF_MF0) % 128;
    const float* W = (li == 0) ? Wm1 : (li == 1) ? Wm2 : (li == 2) ? Wm3 : Wm4;
    int K0, n; decode(local, 4, K0, n);
    a0 = (n < 49 && K0     < 4) ? W[n * 4 + K0]     : 0.0f;
    a1 = (n < 49 && K0 + 1 < 4) ? W[n * 4 + K0 + 1] : 0.0f;
  } else {                                               // fc fragments
    int local = gid - OFF_FCF;
    int K0, n; decode(local, 6, K0, n);
    a0 = fc_elem(fcW, fcb, K0, n);
    a1 = fc_elem(fcW, fcb, K0 + 1, n);
  }
  ws[gid] = make_float2(a0, a1);
}

// ---------------------------------------------------------------------------
// main persistent kernel: 32 WGs x 16 batch rows, 4 waves (one N-tile each)
// ---------------------------------------------------------------------------
__device__ __forceinline__ v8f wmma4(v2f a, v2f b, v8f c) {
  return __builtin_amdgcn_wmma_f32_16x16x4_f32(false, a, false, b,
                                               (short)0, c, false, false);
}

__global__ void __launch_bounds__(128)
lmu_main(const float* __restrict__ x,
         const float* __restrict__ em1, const float* __restrict__ em2,
         const float* __restrict__ em3, const float* __restrict__ em4,
         const v2f* __restrict__ ws, float* __restrict__ out)
{
  __shared__ float xs[16 * XSTR];
  __shared__ float hsB[4][16 * HSTR];
  __shared__ float msB[4][16 * DDm];
  __shared__ float usB[16];
  __shared__ float emS[4][4];

  const int tid   = threadIdx.x;
  const int lane  = tid & 31;
  const int wid   = tid >> 5;             // wave id == output N-tile
  const int row   = lane & 15;            // A-matrix M row / B-matrix N col
  const int kg    = (lane >> 4) << 1;     // A/B lane K offset (0 or 2)
  const int rbase = (lane >> 4) << 3;     // C/D lane M base (0 or 8)
  const int b0    = blockIdx.x * 16;

  // ---- init state ----
  for (int i = tid; i < 4 * 16 * HSTR; i += 128) ((float*)hsB)[i] = 0.0f;
  for (int i = tid; i < 4 * 16 * DDm;  i += 128) ((float*)msB)[i] = 0.0f;
  if (tid < 16) hsB[3][tid * HSTR + 49] = 1.0f;   // bias column for fc
  if (tid < 16) {
    const float* emp = ((tid >> 2) == 0) ? em1 : ((tid >> 2) == 1) ? em2
                     : ((tid >> 2) == 2) ? em3 : em4;
    emS[tid >> 2][tid & 3] = emp[tid & 3];
  }

  // ---- LMU Ad / Bd (exact dyadic values; matches float32 reference) ----
  float Ad[4][4], Bd[4];
  for (int i = 0; i < 4; ++i) {
    float r = (2.0f * (float)i + 1.0f) / 4.0f;
    Bd[i] = (i & 1) ? -r : r;
    for (int j = 0; j < 4; ++j) {
      float Aij = (i < j) ? -r : (((i - j + 1) & 1) ? -r : r);
      Ad[i][j] = ((i == j) ? 1.0f : 0.0f) + Aij;
    }
  }

  const v2f* XF[4] = { ws + OFF_XF0, ws + OFF_XF1, ws + OFF_XF1 + 1664,
                       ws + OFF_XF1 + 3328 };
  const v2f* HF[4] = { ws + OFF_HF0, ws + OFF_HF0 + 1664, ws + OFF_HF0 + 3328,
                       ws + OFF_HF0 + 4992 };
  const v2f* MF[4] = { ws + OFF_MF0, ws + OFF_MF0 + 128, ws + OFF_MF0 + 256,
                       ws + OFF_MF0 + 384 };
  const v2f* FCF   = ws + OFF_FCF;

  // ---- software-pipelined x prefetch: 768 float2 / 128 thr = 6 each ----
  float2 xr[6];
  {
    const float2* xg = (const float2*)(x + (size_t)b0 * INPC);   // t = 0
    #pragma unroll
    for (int j = 0; j < 6; ++j) xr[j] = xg[tid + j * 128];
  }
  __syncthreads();

  for (int t = 0; t < LEN; ++t) {
    // ---- commit prefetched x(t) into LDS ----
    #pragma unroll
    for (int j = 0; j < 6; ++j) {
      int i  = tid + j * 128;
      int fl = i * 2, r = fl / 96, c = fl - r * 96;
      xs[r * XSTR + c]     = xr[j].x;
      xs[r * XSTR + c + 1] = xr[j].y;
    }
    __syncthreads();

    // ---- issue loads for x(t+1); entire step hides their latency ----
    if (t + 1 < LEN) {
      const float2* xg = (const float2*)(x + (size_t)(t + 1) * BSZ * INPC
                                           + (size_t)b0 * INPC);
      #pragma unroll
      for (int j = 0; j < 6; ++j) xr[j] = xg[tid + j * 128];
    }

    for (int l = 0; l < 4; ++l) {
      const float* inb  = (l == 0) ? xs : hsB[l - 1];
      const int    istr = (l == 0) ? XSTR : HSTR;
      const int    kxs  = (l == 0) ? 24 : 13;
      const v2f*   xf   = XF[l];
      const v2f*   hf   = HF[l];

      v8f acc = {0.f, 0.f, 0.f, 0.f, 0.f, 0.f, 0.f, 0.f};

      // input-path matmul (col 49 of fragments carries ex -> u)
      const float* ap = inb + row * istr + kg;
      for (int ks = 0; ks < kxs; ++ks) {
        v2f a; a[0] = ap[ks * 4]; a[1] = ap[ks * 4 + 1];
        v2f b = xf[(ks * 4 + wid) * 32 + lane];
        acc = wmma4(a, b, acc);
      }
      // recurrent matmul (col 49 carries eh -> u)
      const float* hp = hsB[l] + row * HSTR + kg;
      for (int ks = 0; ks < 13; ++ks) {
        v2f a; a[0] = hp[ks * 4]; a[1] = hp[ks * 4 + 1];
        v2f b = hf[(ks * 4 + wid) * 32 + lane];
        acc = wmma4(a, b, acc);
      }

      // extract u (global col 49 == tile 3 local col 1 -> lanes 1 / 17)
      if (wid == 3 && (lane & 15) == 1) {
        for (int v = 0; v < 8; ++v) usB[rbase + v] = acc[v];
      }
      __syncthreads();

      // m update (16 threads, one batch row each) using OLD m
      if (tid < 16) {
        float m0 = msB[l][tid * 4 + 0], m1 = msB[l][tid * 4 + 1];
        float m2 = msB[l][tid * 4 + 2], m3 = msB[l][tid * 4 + 3];
        float u = usB[tid] + emS[l][0] * m0 + emS[l][1] * m1
                           + emS[l][2] * m2 + emS[l][3] * m3;
        float n0 = Ad[0][0]*m0 + Ad[0][1]*m1 + Ad[0][2]*m2 + Ad[0][3]*m3 + u*Bd[0];
        float n1 = Ad[1][0]*m0 + Ad[1][1]*m1 + Ad[1][2]*m2 + Ad[1][3]*m3 + u*Bd[1];
        float n2 = Ad[2][0]*m0 + Ad[2][1]*m1 + Ad[2][2]*m2 + Ad[2][3]*m3 + u*Bd[2];
        float n3 = Ad[3][0]*m0 + Ad[3][1]*m1 + Ad[3][2]*m2 + Ad[3][3]*m3 + u*Bd[3];
        msB[l][tid * 4 + 0] = n0; msB[l][tid * 4 + 1] = n1;
        msB[l][tid * 4 + 2] = n2; msB[l][tid * 4 + 3] = n3;
      }
      __syncthreads();

      // + m_new @ Wm.T : single K=4 WMMA
      {
        v2f a; a[0] = msB[l][row * 4 + kg]; a[1] = msB[l][row * 4 + kg + 1];
        v2f b = MF[l][wid * 32 + lane];
        acc = wmma4(a, b, acc);
      }

      // tanh + write h tile (pad cols: 49 -> bias flag, 50/51 -> 0)
      int col = wid * 16 + row;
      if (col < 49) {
        for (int v = 0; v < 8; ++v)
          hsB[l][(rbase + v) * HSTR + col] = tanhf(acc[v]);
      } else if (col < 52) {
        float val = (col == 49 && l == 3) ? 1.0f : 0.0f;
        for (int v = 0; v < 8; ++v)
          hsB[l][(rbase + v) * HSTR + col] = val;
      }
      __syncthreads();
    }

    // ---- fc: y = h4 @ fcW.T + fcb (bias via h4 col49 == 1) ----
    float* yb = out + (size_t)t * BSZ * INPC + (size_t)b0 * INPC;
    const float* hp4 = hsB[3] + row * HSTR + kg;
    for (int nt = wid; nt < 6; nt += 4) {
      v8f acc = {0.f, 0.f, 0.f, 0.f, 0.f, 0.f, 0.f, 0.f};
      for (int ks = 0; ks < 13; ++ks) {
        v2f a; a[0] = hp4[ks * 4]; a[1] = hp4[ks * 4 + 1];
        v2f b = FCF[(ks * 6 + nt) * 32 + lane];
        acc = wmma4(a, b, acc);
      }
      int col = nt * 16 + row;
      for (int v = 0; v < 8; ++v)
        yb[(size_t)(rbase + v) * INPC + col] = acc[v];
    }
  }

  // ---- final states: (h1,m1,h2,m2,h3,m3,h4,m4) ----
  __syncthreads();
  float* sb = out + (size_t)LEN * BSZ * INPC;
  for (int l = 0; l < 4; ++l) {
    float* hout = sb + (size_t)l * (BSZ * (NN + DDm));
    float* mout = hout + BSZ * NN;
    for (int i = tid; i < 16 * NN; i += 128) {
      int r = i / NN, c = i - r * NN;
      hout[(size_t)(b0 + r) * NN + c] = hsB[l][r * HSTR + c];
    }
    for (int i = tid; i < 16 * DDm; i += 128) {
      int r = i >> 2, c = i & 3;
      mout[(size_t)(b0 + r) * DDm + c] = msB[l][r * DDm + c];
    }
  }
}

// ---------------------------------------------------------------------------
extern "C" void kernel_launch(void* const* d_in, const int* in_sizes, int n_in,
                              void* d_out, int out_size, void* d_ws, size_t ws_size,
                              hipStream_t stream) {
  (void)in_sizes; (void)n_in; (void)out_size; (void)ws_size;
  auto P = [&](int i) { return (const float*)d_in[i]; };
  // d_in: 0:x, then per layer l: ex(1+6l), eh(2+6l), em(3+6l), Wx(4+6l), Wh(5+6l), Wm(6+6l); 25:fcW, 26:fcb
  lmu_prep<<<dim3((TOT_F2 + 127) / 128), dim3(128), 0, stream>>>(
      P(1),  P(2),  P(4),  P(5),  P(6),
      P(7),  P(8),  P(10), P(11), P(12),
      P(13), P(14), P(16), P(17), P(18),
      P(19), P(20), P(22), P(23), P(24),
      P(25), P(26), (float2*)d_ws);

  lmu_main<<<dim3(32), dim3(128), 0, stream>>>(
      P(0), P(3), P(9), P(15), P(21),
      (const v2f*)d_ws, (float*)d_out);
}